// GCN_test_50285477102164
// MI455X (gfx1250) — compile-verified
//
#include <hip/hip_runtime.h>
#include <hip/hip_bf16.h>
#include <math.h>

// ---------------------------------------------------------------------------
// GCN -> fc1 -> fc2 -> conv x4 (WMMA f32 16x16x4 implicit GEMM) -> fc3..fc5
// MI455X (gfx1250, wave32). fc layers are HBM-bandwidth-bound matvecs
// (fc3 = 1.31 GB weights -> split-K for memory-level parallelism);
// convs are compute-shaped GEMMs done on the matrix units in fp32.
// ---------------------------------------------------------------------------

typedef __attribute__((ext_vector_type(2))) float v2f;
typedef __attribute__((ext_vector_type(8))) float v8f;

#define NEG_SLOPE 0.2f

__device__ __forceinline__ float wred(float v) {
#pragma unroll
  for (int o = 16; o > 0; o >>= 1) v += __shfl_xor(v, o, 32);
  return v;
}
__device__ __forceinline__ int wredi(int v) {
#pragma unroll
  for (int o = 16; o > 0; o >>= 1) v += __shfl_xor(v, o, 32);
  return v;
}

// ---------------- GCN stage (deterministic gather, no float atomics) -------

// xw = x @ W1 with x[:, :3] forced to zero (only rows 3..5 of W1 matter)
__global__ __launch_bounds__(256) void k_xw1(const float* __restrict__ x,
                                             const float* __restrict__ W1,
                                             float* __restrict__ xw, int N) {
  int n = blockIdx.x * blockDim.x + threadIdx.x;
  if (n >= N) return;
  float a = x[n * 6 + 3], b = x[n * 6 + 4], c = x[n * 6 + 5];
#pragma unroll
  for (int j = 0; j < 3; ++j)
    xw[n * 3 + j] = a * W1[9 + j] + b * W1[12 + j] + c * W1[15 + j];
}

// xw = h @ W2   (3x3)
__global__ __launch_bounds__(256) void k_xw2(const float* __restrict__ h,
                                             const float* __restrict__ W2,
                                             float* __restrict__ xw, int N) {
  int n = blockIdx.x * blockDim.x + threadIdx.x;
  if (n >= N) return;
  float a = h[n * 3 + 0], b = h[n * 3 + 1], c = h[n * 3 + 2];
#pragma unroll
  for (int j = 0; j < 3; ++j)
    xw[n * 3 + j] = a * W2[j] + b * W2[3 + j] + c * W2[6 + j];
}

// dinv[n] = (deg_in(n) + 1)^-1/2   (self-loop included); wave per node
__global__ __launch_bounds__(256) void k_deg(const int* __restrict__ dst,
                                             float* __restrict__ dinv, int N,
                                             int E) {
  int n = blockIdx.x * 8 + threadIdx.y;
  if (n >= N) return;
  int cnt = 0;
  for (int e = threadIdx.x; e < E; e += 32) cnt += (dst[e] == n) ? 1 : 0;
  cnt = wredi(cnt);
  if (threadIdx.x == 0) dinv[n] = rsqrtf((float)(cnt + 1));
}

// h[n] = leaky( sum_{e: dst==n} dinv[src]*dinv[n]*xw[src] + dinv[n]^2*xw[n] + b )
__global__ __launch_bounds__(256) void k_gather(
    const float* __restrict__ xw, const float* __restrict__ dinv,
    const int* __restrict__ src, const int* __restrict__ dst,
    const float* __restrict__ bias, float* __restrict__ h, int N, int E) {
  int n = blockIdx.x * 8 + threadIdx.y;
  if (n >= N) return;
  float di = dinv[n];
  float a0 = 0.f, a1 = 0.f, a2 = 0.f;
  for (int e = threadIdx.x; e < E; e += 32) {
    if (dst[e] == n) {
      int s = src[e];
      float f = dinv[s] * di;
      a0 += f * xw[s * 3 + 0];
      a1 += f * xw[s * 3 + 1];
      a2 += f * xw[s * 3 + 2];
    }
  }
  a0 = wred(a0); a1 = wred(a1); a2 = wred(a2);
  if (threadIdx.x == 0) {
    float fs = di * di;
    float r0 = a0 + fs * xw[n * 3 + 0] + bias[0];
    float r1 = a1 + fs * xw[n * 3 + 1] + bias[1];
    float r2 = a2 + fs * xw[n * 3 + 2] + bias[2];
    h[n * 3 + 0] = r0 >= 0.f ? r0 : NEG_SLOPE * r0;
    h[n * 3 + 1] = r1 >= 0.f ? r1 : NEG_SLOPE * r1;
    h[n * 3 + 2] = r2 >= 0.f ? r2 : NEG_SLOPE * r2;
  }
}

// ---------------- Bandwidth-optimal matvecs: out = W @ v + b ---------------

// generic (K any): one wave per row, coalesced b32 loads
__global__ __launch_bounds__(256) void matvec1(const float* __restrict__ W,
                                               const float* __restrict__ v,
                                               const float* __restrict__ b,
                                               float* __restrict__ out, int M,
                                               int K) {
  int row = blockIdx.x * 8 + threadIdx.y;
  if (row >= M) return;
  const float* wr = W + (size_t)row * K;
  float acc = 0.f;
  for (int k = threadIdx.x; k < K; k += 32) acc += wr[k] * v[k];
  acc = wred(acc);
  if (threadIdx.x == 0) out[row] = acc + b[row];
}

// K % 4 == 0: float4 streaming (b128 loads), one wave per row
__global__ __launch_bounds__(256) void matvec4(const float* __restrict__ W,
                                               const float* __restrict__ v,
                                               const float* __restrict__ b,
                                               float* __restrict__ out, int M,
                                               int K4) {
  int row = blockIdx.x * 8 + threadIdx.y;
  if (row >= M) return;
  const float4* wr = (const float4*)W + (size_t)row * K4;
  const float4* v4 = (const float4*)v;
  float acc = 0.f;
  for (int k = threadIdx.x; k < K4; k += 32) {
    __builtin_prefetch(wr + k + 256, 0, 0);  // global_prefetch_b8, stream ahead
    float4 a = wr[k];
    float4 x = v4[k];
    acc += a.x * x.x + a.y * x.y + a.z * x.z + a.w * x.w;
  }
  acc = wred(acc);
  if (threadIdx.x == 0) out[row] = acc + b[row];
}

// Split-K partial matvec for the 1.31 GB fc3 layer: grid.y = chunk index.
// Deterministic: fixed partition, fixed in-wave reduction order.
template <int S>
__global__ __launch_bounds__(256) void matvec4_part(
    const float* __restrict__ W, const float* __restrict__ v,
    float* __restrict__ partial, int M, int K4) {
  int row = blockIdx.x * 8 + threadIdx.y;
  if (row >= M) return;
  const int Kc = K4 / S;               // chunk length (in float4s)
  const int s = blockIdx.y;
  const float4* wr = (const float4*)W + (size_t)row * K4 + (size_t)s * Kc;
  const float4* v4 = (const float4*)v + (size_t)s * Kc;
  float acc = 0.f;
  for (int k = threadIdx.x; k < Kc; k += 32) {
    __builtin_prefetch(wr + k + 256, 0, 0);
    float4 a = wr[k];
    float4 x = v4[k];
    acc += a.x * x.x + a.y * x.y + a.z * x.z + a.w * x.w;
  }
  acc = wred(acc);
  if (threadIdx.x == 0) partial[(size_t)row * S + s] = acc;
}

template <int S>
__global__ __launch_bounds__(256) void k_part_reduce(
    const float* __restrict__ partial, const float* __restrict__ b,
    float* __restrict__ out, int M) {
  int row = blockIdx.x * blockDim.x + threadIdx.x;
  if (row >= M) return;
  float acc = b[row];
#pragma unroll
  for (int s = 0; s < S; ++s) acc += partial[(size_t)row * S + s];
  out[row] = acc;
}

// ---------------- Convolution via V_WMMA_F32_16X16X4_F32 -------------------
// Implicit GEMM: D[M=Cout][N=Ho*Wo] = W[M][K=Ci*kh*kw] x Im2col[K][N].
// One wave owns a 16x16 output tile. fp32 A/B keeps reference numerics.
// A 16x4 layout:  lane = M + 16*(K>>1), vgpr = K&1   (ISA 7.12.2)
// B 4x16 layout:  vgpr r, lanes 0-15: K=r, N=lane; lanes 16-31: K=r+2
// C/D 16x16:      vgpr r, lanes 0-15: M=r;  lanes 16-31: M=r+8; N=lane&15
template <int O, int I, int KH, int KW, int HIN, int WIN>
__global__ __launch_bounds__(128) void conv_wmma(const float* __restrict__ in,
                                                 const float* __restrict__ w,
                                                 const float* __restrict__ bias,
                                                 float* __restrict__ out) {
  constexpr int HOUT = HIN - KH + 1;
  constexpr int WOUT = WIN - KW + 1;
  constexpr int NT = HOUT * WOUT;
  constexpr int K = I * KH * KW;

  __shared__ int offs[K];  // k -> input offset of (ci, kh, kw) at (y=0,x=0)
  int tid = threadIdx.y * 32 + threadIdx.x;
  for (int k = tid; k < K; k += 128) {
    int ci = k / (KH * KW);
    int r = k % (KH * KW);
    offs[k] = ci * (HIN * WIN) + (r / KW) * WIN + (r % KW);
  }
  __syncthreads();

  int lane = threadIdx.x;
  int l15 = lane & 15;
  int lhi = lane >> 4;
  int n0 = (blockIdx.x * 4 + threadIdx.y) * 16;
  int m0 = blockIdx.y * 16;

  int nIdx = n0 + l15;
  int nn = (nIdx < NT) ? nIdx : 0;          // clamp; garbage cols never stored
  int bbase = (nn / WOUT) * WIN + (nn % WOUT);
  const float* wrow = w + (size_t)(m0 + l15) * K;
  int kv = lhi * 2;                          // this lane's K sub-index (0 or 2)

  v8f c = {0.f, 0.f, 0.f, 0.f, 0.f, 0.f, 0.f, 0.f};
  for (int k0 = 0; k0 < K; k0 += 4) {
    int ka = k0 + kv, kb = ka + 1;
    float m0f = (ka < K) ? 1.f : 0.f;        // K-tail masks (selects, EXEC stays full)
    float m1f = (kb < K) ? 1.f : 0.f;
    int kac = (ka < K) ? ka : 0;
    int kbc = (kb < K) ? kb : 0;
    v2f a = {wrow[kac] * m0f, wrow[kbc] * m1f};
    v2f bm = {in[offs[kac] + bbase] * m0f, in[offs[kbc] + bbase] * m1f};
    c = __builtin_amdgcn_wmma_f32_16x16x4_f32(false, a, false, bm, (short)0, c,
                                              false, false);
  }

  if (nIdx < NT) {
#pragma unroll
    for (int r = 0; r < 8; ++r) {
      int m = m0 + r + 8 * lhi;
      out[(size_t)m * NT + nIdx] = c[r] + bias[m];
    }
  }
}

// ---------------------------------------------------------------------------

extern "C" void kernel_launch(void* const* d_in, const int* in_sizes, int n_in,
                              void* d_out, int out_size, void* d_ws,
                              size_t ws_size, hipStream_t stream) {
  const float* x = (const float*)d_in[0];
  const int* ei = (const int*)d_in[1];
  const int N = in_sizes[0] / 6;
  const int E = in_sizes[1] / 2;
  const int* esrc = ei;
  const int* edst = ei + E;
  const float* W1 = (const float*)d_in[2];
  const float* b1 = (const float*)d_in[3];
  const float* W2 = (const float*)d_in[4];
  const float* b2 = (const float*)d_in[5];
  const float* fc1_w = (const float*)d_in[6];
  const float* fc1_b = (const float*)d_in[7];
  const float* fc2_w = (const float*)d_in[8];
  const float* fc2_b = (const float*)d_in[9];
  const float* c1_w = (const float*)d_in[10];
  const float* c1_b = (const float*)d_in[11];
  const float* c2_w = (const float*)d_in[12];
  const float* c2_b = (const float*)d_in[13];
  const float* c3_w = (const float*)d_in[14];
  const float* c3_b = (const float*)d_in[15];
  const float* c4_w = (const float*)d_in[16];
  const float* c4_b = (const float*)d_in[17];
  const float* fc3_w = (const float*)d_in[18];
  const float* fc3_b = (const float*)d_in[19];
  const float* fc4_w = (const float*)d_in[20];
  const float* fc4_b = (const float*)d_in[21];
  const float* fc5_w = (const float*)d_in[22];
  const float* fc5_b = (const float*)d_in[23];

  // workspace layout (floats, 16B-aligned offsets for float4 paths)
  float* ws = (float*)d_ws;
  float* dinv = ws + 0;       // 1395
  float* xw = ws + 2048;      // 4185
  float* h = ws + 8192;       // 4185  (== v[4185] after layer 2)
  float* v1 = ws + 16384;     // 2048
  float* v2 = ws + 20480;     // 4096  (== img 64x64)
  float* c1o = ws + 32768;    // 32*58*58 = 107648
  float* c2o = ws + 163840;   // 32*54*54 =  93312
  float* c3o = ws + 262144;   // 64*52*52 = 173056
  float* c4o = ws + 442368;   // 64*50*50 = 160000
  float* v3 = ws + 606208;    // 2048
  float* v4 = ws + 610304;    // 2048
  float* part = ws + 614400;  // 2048*16 split-K partials for fc3

  dim3 w8(32, 8);
  dim3 cblk(32, 4);

  // --- GCN layer 1 ---
  k_xw1<<<(N + 255) / 256, 256, 0, stream>>>(x, W1, xw, N);
  k_deg<<<(N + 7) / 8, w8, 0, stream>>>(edst, dinv, N, E);
  k_gather<<<(N + 7) / 8, w8, 0, stream>>>(xw, dinv, esrc, edst, b1, h, N, E);
  // --- GCN layer 2 ---
  k_xw2<<<(N + 255) / 256, 256, 0, stream>>>(h, W2, xw, N);
  k_gather<<<(N + 7) / 8, w8, 0, stream>>>(xw, dinv, esrc, edst, b2, h, N, E);

  // --- fc1: 2048 x 4185 (odd K -> scalar path), fc2: 4096 x 2048 ---
  matvec1<<<2048 / 8, w8, 0, stream>>>(fc1_w, h, fc1_b, v1, 2048, 4185);
  matvec4<<<4096 / 8, w8, 0, stream>>>(fc2_w, v1, fc2_b, v2, 4096, 2048 / 4);

  // --- convolutions on the matrix units (fp32 WMMA) ---
  {  // conv1: 1x64x64 -> 32x58x58, 7x7  (NT=3364)
    dim3 g((3364 + 63) / 64, 32 / 16);
    conv_wmma<32, 1, 7, 7, 64, 64><<<g, cblk, 0, stream>>>(v2, c1_w, c1_b, c1o);
  }
  {  // conv2: 32x58x58 -> 32x54x54, 5x5 (NT=2916, K=800)
    dim3 g((2916 + 63) / 64, 32 / 16);
    conv_wmma<32, 32, 5, 5, 58, 58><<<g, cblk, 0, stream>>>(c1o, c2_w, c2_b, c2o);
  }
  {  // conv3: 32x54x54 -> 64x52x52, 3x3 (NT=2704, K=288)
    dim3 g((2704 + 63) / 64, 64 / 16);
    conv_wmma<64, 32, 3, 3, 54, 54><<<g, cblk, 0, stream>>>(c2o, c3_w, c3_b, c3o);
  }
  {  // conv4: 64x52x52 -> 64x50x50, 3x3 (NT=2500, K=576)
    dim3 g((2500 + 63) / 64, 64 / 16);
    conv_wmma<64, 64, 3, 3, 52, 52><<<g, cblk, 0, stream>>>(c3o, c4_w, c4_b, c4o);
  }

  // --- fc3 (HBM-bound: 1.31 GB of weights) -> split-K x16 for MLP of waves,
  //     then deterministic reduction; fc4, fc5 as plain matvecs ---
  {
    dim3 g(2048 / 8, 16);
    matvec4_part<16><<<g, w8, 0, stream>>>(fc3_w, c4o, part, 2048, 160000 / 4);
    k_part_reduce<16><<<(2048 + 255) / 256, 256, 0, stream>>>(part, fc3_b, v3,
                                                              2048);
  }
  matvec4<<<2048 / 8, w8, 0, stream>>>(fc4_w, v3, fc4_b, v4, 2048, 2048 / 4);
  matvec4<<<48 / 8, w8, 0, stream>>>(fc5_w, v4, fc5_b, (float*)d_out, 48,
                                     2048 / 4);
}